// RelationMultiHeadAttention_77644418777126
// MI455X (gfx1250) — compile-verified
//
#include <hip/hip_runtime.h>
#include <math.h>

typedef __attribute__((ext_vector_type(2))) float v2f;
typedef __attribute__((ext_vector_type(8))) float v8f;

#define NN      50000
#define NUSERS  20000
#define EE      800000
#define DIM     256
#define HH      8
#define DKK     32
#define LDS_PITCH 260   // 256 + 4 pad: row stride 1040B (16B aligned), banks 0,4,..,60

// ---------------------------------------------------------------------------
// WMMA fp32 GEMM:  out[r, c] = sum_k src[r, k] * W[c, k] + bias[c]
// src rows come from srcA for r < split, else srcB (rebased) — handles the
// concat(user_text_embs, item_text_embs) without materializing text_emb.
// Block = 8 waves; block owns 16 rows x 256 cols. A tile staged in LDS once
// (coalesced), each wave computes two 16x16 tiles (32 cols) from LDS A.
// grid = (NN/16).
// ---------------------------------------------------------------------------
__global__ __launch_bounds__(256) void gemm16_wmma(
    const float* __restrict__ srcA, const float* __restrict__ srcB, int split,
    const float* __restrict__ W, const float* __restrict__ bias,
    float* __restrict__ out)
{
    __shared__ float As[16 * LDS_PITCH];

    const int tid  = threadIdx.x;
    const int row0 = blockIdx.x * 16;

    // ---- cooperative A-tile load: 16 rows x 256 floats = 1024 float4s ----
    #pragma unroll
    for (int i = 0; i < 4; ++i) {
        const int idx4 = tid + i * 256;       // float4 index 0..1023
        const int r    = idx4 >> 6;           // row within tile
        const int c4   = idx4 & 63;           // float4 within row
        const int gr   = row0 + r;
        const float* src = (gr < split) ? (srcA + (size_t)gr * DIM)
                                        : (srcB + (size_t)(gr - split) * DIM);
        const float4 v = *(const float4*)(src + c4 * 4);
        *(float4*)(&As[r * LDS_PITCH + c4 * 4]) = v;
    }
    __syncthreads();

    // ---- per-wave compute: two 16x16 output tiles ----
    const int wave = tid >> 5;
    const int lane = tid & 31;
    const int lr   = lane & 15;     // row (A) / col (B) within tile
    const int half = lane >> 4;     // K sub-pair selector

    const int col0 = wave * 32;
    const float* arow = &As[lr * LDS_PITCH];
    const float* w0   = W + (size_t)(col0 + lr) * DIM;       // B[k][n] = W[n][k]
    const float* w1   = W + (size_t)(col0 + 16 + lr) * DIM;

    v8f acc0 = {};
    v8f acc1 = {};
    #pragma unroll 4
    for (int k0 = 0; k0 < DIM; k0 += 4) {
        // A 16x4: lane<16 -> K=k0..k0+1 ; lane>=16 -> K=k0+2..k0+3
        v2f a  = *(const v2f*)(arow + k0 + 2 * half);
        v2f b0 = *(const v2f*)(w0 + k0 + 2 * half);
        v2f b1 = *(const v2f*)(w1 + k0 + 2 * half);
        acc0 = __builtin_amdgcn_wmma_f32_16x16x4_f32(
            false, a, false, b0, (short)0, acc0, false, false);
        acc1 = __builtin_amdgcn_wmma_f32_16x16x4_f32(
            false, a, false, b1, (short)0, acc1, false, false);
    }

    const int c0 = col0 + lr;
    const int c1 = col0 + 16 + lr;
    const float bb0 = bias[c0];
    const float bb1 = bias[c1];
    #pragma unroll
    for (int v = 0; v < 8; ++v) {
        const int rr = row0 + v + 8 * half;   // C: VGPR v -> M=v (lanes<16) / v+8
        out[(size_t)rr * DIM + c0] = acc0[v] + bb0;
        out[(size_t)rr * DIM + c1] = acc1[v] + bb1;
    }
}

// ---------------------------------------------------------------------------
// Init: segment-max to -inf, denominators and output accumulator to 0.
// ---------------------------------------------------------------------------
__global__ void init_buffers(float* __restrict__ mbuf, float* __restrict__ denom,
                             float* __restrict__ acc)
{
    const size_t t = (size_t)blockIdx.x * blockDim.x + threadIdx.x;
    if (t < (size_t)NN * HH) {
        mbuf[t]  = -__builtin_inff();
        denom[t] = 0.0f;
    }
    if (t < (size_t)NN * DIM) acc[t] = 0.0f;
}

// ---------------------------------------------------------------------------
// Pass A: scores[e,h] = dot(Q[col,h,:], K[row,h,:] + rel[h,:]) / sqrt(32)
// plus running segment max over destination node via float atomicMax.
// One thread per (edge, head).  Q/K are L2-resident (51MB each, 192MB L2).
// ---------------------------------------------------------------------------
__global__ __launch_bounds__(256) void edge_scores(
    const float* __restrict__ Q, const float* __restrict__ Kx,
    const float* __restrict__ rel, const int* __restrict__ row,
    const int* __restrict__ col, float* __restrict__ scores,
    float* __restrict__ mbuf)
{
    const int t = blockIdx.x * 256 + threadIdx.x;
    const int e = t >> 3;
    const int h = t & 7;
    if (e >= EE) return;

    const int rs = row[e];
    const int cd = col[e];
    const float* q  = Q  + (size_t)cd * DIM + h * DKK;
    const float* k  = Kx + (size_t)rs * DIM + h * DKK;
    const float* rp = rel + h * DKK;

    float s = 0.0f;
    #pragma unroll
    for (int i = 0; i < DKK; i += 4) {
        float4 qv = *(const float4*)(q + i);
        float4 kv = *(const float4*)(k + i);
        float4 rv = *(const float4*)(rp + i);
        s += qv.x * (kv.x + rv.x) + qv.y * (kv.y + rv.y)
           + qv.z * (kv.z + rv.z) + qv.w * (kv.w + rv.w);
    }
    s *= 0.17677669529663687f;  // 1/sqrt(32)

    scores[t] = s;
    atomicMax(mbuf + (size_t)cd * HH + h, s);
}

// ---------------------------------------------------------------------------
// Pass B (fused): ex = exp(score - m[col]);
//   denom[col,h] += ex;   acc[col,h,:] += ex * V[row,h,:]   (unnormalized)
// Normalization is deferred to a per-node pass: sum(ex*V)/denom == sum(attn*V).
// ---------------------------------------------------------------------------
__global__ __launch_bounds__(256) void edge_expagg(
    const int* __restrict__ row, const int* __restrict__ col,
    const float* __restrict__ scores, const float* __restrict__ mbuf,
    const float* __restrict__ V, float* __restrict__ denom,
    float* __restrict__ acc)
{
    const int t = blockIdx.x * 256 + threadIdx.x;
    const int e = t >> 3;
    const int h = t & 7;
    if (e >= EE) return;

    const int rs = row[e];
    const int cd = col[e];
    const float m  = mbuf[(size_t)cd * HH + h];
    const float ex = expf(scores[t] - m);

    atomicAdd(denom + (size_t)cd * HH + h, ex);

    const float* v = V + (size_t)rs * DIM + h * DKK;
    float* a = acc + (size_t)cd * DIM + h * DKK;
    #pragma unroll
    for (int i = 0; i < DKK; ++i)
        atomicAdd(a + i, ex * v[i]);
}

// ---------------------------------------------------------------------------
// Per-node normalize: acc[n, h, :] /= denom[n, h]  (guard isolated nodes).
// One thread per output element.
// ---------------------------------------------------------------------------
__global__ __launch_bounds__(256) void node_normalize(
    const float* __restrict__ denom, float* __restrict__ acc)
{
    const int t = blockIdx.x * 256 + threadIdx.x;  // n*DIM + c
    const int n = t >> 8;
    const int c = t & 255;
    if (n >= NN) return;
    const float d = denom[(size_t)n * HH + (c >> 5)];
    acc[t] = (d > 0.0f) ? (acc[t] / d) : 0.0f;
}

// ---------------------------------------------------------------------------
// Launch
// ---------------------------------------------------------------------------
extern "C" void kernel_launch(void* const* d_in, const int* in_sizes, int n_in,
                              void* d_out, int out_size, void* d_ws, size_t ws_size,
                              hipStream_t stream)
{
    const float* x    = (const float*)d_in[0];
    const int*   row  = (const int*)  d_in[1];
    const int*   col  = (const int*)  d_in[2];
    const float* rel  = (const float*)d_in[3];
    const float* uemb = (const float*)d_in[4];
    const float* iemb = (const float*)d_in[5];
    const float* Wq   = (const float*)d_in[6];
    const float* bq   = (const float*)d_in[7];
    const float* Wk   = (const float*)d_in[8];
    const float* bk   = (const float*)d_in[9];
    const float* Wv   = (const float*)d_in[10];
    const float* bv   = (const float*)d_in[11];
    const float* Wo   = (const float*)d_in[12];
    const float* bo   = (const float*)d_in[13];
    float* out = (float*)d_out;

    float* ws     = (float*)d_ws;
    float* Q      = ws;                              // N*DIM
    float* K      = Q   + (size_t)NN * DIM;          // N*DIM
    float* V      = K   + (size_t)NN * DIM;          // N*DIM
    float* acc    = V   + (size_t)NN * DIM;          // N*DIM
    float* scores = acc + (size_t)NN * DIM;          // E*H
    float* mbuf   = scores + (size_t)EE * HH;        // N*H
    float* denom  = mbuf   + (size_t)NN * HH;        // N*H

    const dim3 gg(NN / 16);
    const dim3 bb(256);

    gemm16_wmma<<<gg, bb, 0, stream>>>(uemb, iemb, NUSERS, Wq, bq, Q);
    gemm16_wmma<<<gg, bb, 0, stream>>>(uemb, iemb, NUSERS, Wk, bk, K);
    gemm16_wmma<<<gg, bb, 0, stream>>>(x,    x,    NN,     Wv, bv, V);

    init_buffers<<<(NN * DIM + 255) / 256, 256, 0, stream>>>(mbuf, denom, acc);

    const int edge_blocks = (EE * HH) / 256;   // 25000, exact
    edge_scores<<<edge_blocks, 256, 0, stream>>>(Q, K, rel, row, col, scores, mbuf);
    edge_expagg<<<edge_blocks, 256, 0, stream>>>(row, col, scores, mbuf, V, denom, acc);

    node_normalize<<<(NN * DIM) / 256, 256, 0, stream>>>(denom, acc);

    gemm16_wmma<<<gg, bb, 0, stream>>>(acc, acc, NN, Wo, bo, out);
}